// FrequencyToSpatialCrossAttention_22935125360697
// MI455X (gfx1250) — compile-verified
//
#include <hip/hip_runtime.h>
#include <math.h>

// Problem constants (match reference)
#define B_      4
#define C_      384
#define H_      256
#define W_      256
#define NH      8
#define HD      48          // C_/NH
#define WS      8
#define NWIN    32          // H_/WS per dim
#define NW      (NWIN*NWIN) // 1024 windows
#define SHIFT_  4

typedef __attribute__((ext_vector_type(2))) float v2f;
typedef __attribute__((ext_vector_type(8))) float v8f;

// ---------------- workspace layout (floats) ----------------
#define WS_QG   0                       // 16 x 384 (rows 4..15 stay zero for WMMA padding)
#define WS_Q    (WS_QG  + 16*C_)        // 4 x 384
#define WS_QK   (WS_Q   + 4*C_)         // 4 x 8 x 384 (scale folded in)
#define WS_QB   (WS_QK  + B_*NH*C_)     // 32
#define WS_Y    (WS_QB  + B_*NH)        // 4 x 8 x 384 accumulators
#define WS_MID  (WS_Y   + B_*NH*C_)     // 16 x 384 (rows 4..15 zero)
#define WS_FIN  (WS_MID + 16*C_)        // 4 x 384
#define WS_TOT  (WS_FIN + 4*C_)

// ---------------- K0: zero workspace (needed every call: y is accumulated) ----
__global__ void zero_ws_kernel(float* ws, int n) {
    int i = blockIdx.x * 256 + threadIdx.x;
    if (i < n) ws[i] = 0.0f;
}

// ---------------- K1: Qg[b,c] = mean over HxW of X_f -------------------------
__global__ void mean_kernel(const float* __restrict__ Xf, float* __restrict__ qg) {
    __shared__ float red[256];
    const int bc = blockIdx.x;                       // b*C_ + c
    const float* p = Xf + (size_t)bc * (H_ * W_);
    float s = 0.0f;
    for (int i = threadIdx.x; i < H_ * W_; i += 256) s += p[i];
    red[threadIdx.x] = s;
    __syncthreads();
    for (int off = 128; off > 0; off >>= 1) {
        if (threadIdx.x < off) red[threadIdx.x] += red[threadIdx.x + off];
        __syncthreads();
    }
    if (threadIdx.x == 0) qg[bc] = red[0] * (1.0f / (H_ * W_));
}

// ---------------- K2/K5b: 16xC @ CxC GEMM via V_WMMA_F32_16X16X4_F32 ---------
// D[m,n] = sum_k A[m,k] * Wmat[n,k] + bias[n]; only rows 0..3 stored.
// 128 threads = 4 waves; each wave owns 6 of the 24 N-tiles.
__global__ __launch_bounds__(128) void proj16_wmma_kernel(
        const float* __restrict__ A,      // 16 x C_ row-major (rows >=4 are zero)
        const float* __restrict__ Wmat,   // C_ x C_ ; B-matrix[k][n] = Wmat[n*C_+k]
        const float* __restrict__ bias,
        float* __restrict__ out)          // 4 x C_
{
    const int lane = threadIdx.x & 31;
    const int wave = threadIdx.x >> 5;
    const int m    = lane & 15;           // M (A) / N (B) lane position
    const int koff = (lane >> 4) << 1;    // upper half-wave handles K+2,K+3

    for (int nt = wave; nt < C_ / 16; nt += 4) {
        const int n0 = nt * 16;
        v8f acc = {};
        for (int k = 0; k < C_; k += 4) {
            const int kk = k + koff;
            v2f a, b;
            // A 16x4 layout: VGPR0 -> K=kk, VGPR1 -> K=kk+1
            a.x = A[m * C_ + kk];
            a.y = A[m * C_ + kk + 1];
            // B 4x16 layout: VGPR0 -> (K=kk, N=n0+m), VGPR1 -> K=kk+1
            b.x = Wmat[(n0 + m) * C_ + kk];
            b.y = Wmat[(n0 + m) * C_ + kk + 1];
            acc = __builtin_amdgcn_wmma_f32_16x16x4_f32(
                      false, a, false, b, (short)0, acc, false, false);
        }
        // C/D layout: VGPR r holds M=r for lanes 0..15 (M=r+8 for 16..31)
        if (lane < 16) {
            #pragma unroll
            for (int r = 0; r < 4; ++r)
                out[r * C_ + n0 + lane] = acc[r] + bias[n0 + lane];
        }
    }
}

// ---------------- K3: fold Wk (and scale) into the query ---------------------
// qk[b,h,c] = scale * sum_d Q[b,h*HD+d] * Wk[(h*HD+d)*C_+c];  qb likewise with bk
__global__ void fold_qk_kernel(const float* __restrict__ q,
                               const float* __restrict__ Wk,
                               const float* __restrict__ bk,
                               float* __restrict__ qk,
                               float* __restrict__ qb) {
    const int bh = blockIdx.x;            // b*NH + h
    const int b = bh >> 3, h = bh & 7;
    const float scale = rsqrtf((float)HD);
    const int c = threadIdx.x;            // 384 threads
    float acc = 0.0f;
    for (int d = 0; d < HD; ++d)
        acc += q[b * C_ + h * HD + d] * Wk[(h * HD + d) * C_ + c];
    qk[bh * C_ + c] = acc * scale;
    if (c == 0) {
        float s = 0.0f;
        for (int d = 0; d < HD; ++d)
            s += q[b * C_ + h * HD + d] * bk[h * HD + d];
        qb[bh] = s * scale;
    }
}

// ---------------- K4: per-window scores + softmax + weighted-x accumulation --
// One 64-thread block per (window, batch). Window (64 pix x 384 ch) staged in
// LDS with stride-65 padding (bank-conflict-free in both access phases).
#define XS_STRIDE 65
#define SM_QS   0                         // 8*384   = 3072
#define SM_XS   (SM_QS + NH*C_)           // 384*65  = 24960
#define SM_S    (SM_XS + C_*XS_STRIDE)    // 8*64    = 512
#define SM_M    (SM_S + NH*64)            // 8
#define SM_I    (SM_M + NH)               // 8
#define SM_A    (SM_I + NH)               // 8*64    = 512
#define SM_TOT  (SM_A + NH*64)

__global__ __launch_bounds__(64) void attn_window_kernel(
        const float* __restrict__ Xs,
        const float* __restrict__ qkg,
        const float* __restrict__ qbg,
        float* __restrict__ y) {
    extern __shared__ float sm[];
    float* qs   = sm + SM_QS;
    float* xs   = sm + SM_XS;
    float* S    = sm + SM_S;
    float* mbuf = sm + SM_M;
    float* ibuf = sm + SM_I;
    float* Abuf = sm + SM_A;

    const int b  = blockIdx.y;
    const int w  = blockIdx.x;
    const int wi = w >> 5, wj = w & 31;
    const int t  = threadIdx.x;           // 0..63

    // preload folded query for this batch
    for (int i = t; i < NH * C_; i += 64) qs[i] = qkg[b * NH * C_ + i];
    __syncthreads();

    // ---- phase 1: thread = pixel; stream channels (coalesced), stage in LDS,
    //      and accumulate the 8 head scores on the fly.
    const int r  = t >> 3, cq = t & 7;
    const int gi = (wi * WS + r  - SHIFT_) & (H_ - 1);   // cyclic shift (roll)
    const int gj = (wj * WS + cq - SHIFT_) & (W_ - 1);
    const float* xp = Xs + ((size_t)(b * C_) * H_ + gi) * W_ + gj;

    float acc[NH];
    #pragma unroll
    for (int h = 0; h < NH; ++h) acc[h] = 0.0f;

    for (int c = 0; c < C_; ++c) {
        const float v = xp[(size_t)c * (H_ * W_)];
        xs[c * XS_STRIDE + t] = v;
        #pragma unroll
        for (int h = 0; h < NH; ++h) acc[h] += v * qs[h * C_ + c];
    }
    #pragma unroll
    for (int h = 0; h < NH; ++h) S[h * 64 + t] = acc[h] + qbg[b * NH + h];
    __syncthreads();

    // ---- phase 2: softmax over the 64 window positions, per head
    if (t < NH) {
        float mx = -INFINITY;
        for (int p = 0; p < 64; ++p) mx = fmaxf(mx, S[t * 64 + p]);
        float ssum = 0.0f;
        for (int p = 0; p < 64; ++p) ssum += __expf(S[t * 64 + p] - mx);
        mbuf[t] = mx;
        ibuf[t] = 1.0f / ssum;
    }
    __syncthreads();
    #pragma unroll
    for (int h = 0; h < NH; ++h)
        Abuf[h * 64 + t] = __expf(S[h * 64 + t] - mbuf[h]) * ibuf[h];
    __syncthreads();

    // ---- phase 3: thread = channel (6 per thread); y[b,h,c] += sum_p A*x
    float ya[6][NH];
    #pragma unroll
    for (int ci = 0; ci < 6; ++ci)
        #pragma unroll
        for (int h = 0; h < NH; ++h) ya[ci][h] = 0.0f;

    for (int p = 0; p < 64; ++p) {
        float aw[NH];
        #pragma unroll
        for (int h = 0; h < NH; ++h) aw[h] = Abuf[h * 64 + p];   // broadcast
        #pragma unroll
        for (int ci = 0; ci < 6; ++ci) {
            const float xv = xs[(t + 64 * ci) * XS_STRIDE + p];  // conflict-free
            #pragma unroll
            for (int h = 0; h < NH; ++h) ya[ci][h] += xv * aw[h];
        }
    }
    for (int ci = 0; ci < 6; ++ci) {
        const int c = t + 64 * ci;
        #pragma unroll
        for (int h = 0; h < NH; ++h)
            atomicAdd(&y[(b * NH + h) * C_ + c], ya[ci][h]);
    }
}

// ---------------- K5a: apply Wv (block-diagonal per head) + window mean ------
__global__ void vproj_kernel(const float* __restrict__ y,
                             const float* __restrict__ Wv,
                             const float* __restrict__ bv,
                             float* __restrict__ mid) {
    const int idx = blockIdx.x * 256 + threadIdx.x;
    if (idx >= B_ * C_) return;
    const int b = idx / C_, n = idx % C_;
    const int h = n / HD;
    float acc = 0.0f;
    for (int c = 0; c < C_; ++c)
        acc += y[(b * NH + h) * C_ + c] * Wv[n * C_ + c];
    mid[b * C_ + n] = acc * (1.0f / NW) + bv[n];
}

// ---------------- K6: broadcast (B,C) -> (B,C,H,W) ---------------------------
__global__ void bcast_kernel(const float* __restrict__ fin, float* __restrict__ out) {
    const int bc = blockIdx.x;
    const float v = fin[bc];
    float4 v4 = make_float4(v, v, v, v);
    float4* o = (float4*)(out + (size_t)bc * (H_ * W_));
    for (int i = threadIdx.x; i < (H_ * W_) / 4; i += blockDim.x) o[i] = v4;
}

// ---------------- launch ------------------------------------------------------
extern "C" void kernel_launch(void* const* d_in, const int* in_sizes, int n_in,
                              void* d_out, int out_size, void* d_ws, size_t ws_size,
                              hipStream_t stream) {
    const float* Xf = (const float*)d_in[0];
    const float* Xs = (const float*)d_in[1];
    const float* Wq = (const float*)d_in[2];
    const float* bq = (const float*)d_in[3];
    const float* Wk = (const float*)d_in[4];
    const float* bk = (const float*)d_in[5];
    const float* Wv = (const float*)d_in[6];
    const float* bv = (const float*)d_in[7];
    const float* Wo = (const float*)d_in[8];
    const float* bo = (const float*)d_in[9];
    float* out = (float*)d_out;

    float* ws  = (float*)d_ws;
    float* qg  = ws + WS_QG;
    float* q   = ws + WS_Q;
    float* qk  = ws + WS_QK;
    float* qb  = ws + WS_QB;
    float* y   = ws + WS_Y;
    float* mid = ws + WS_MID;
    float* fin = ws + WS_FIN;

    // K0: zero workspace (y accumulators + WMMA zero-padding rows)
    zero_ws_kernel<<<(WS_TOT + 255) / 256, 256, 0, stream>>>(ws, WS_TOT);

    // K1: global-pool of X_f -> Qg (403 MB stream)
    mean_kernel<<<B_ * C_, 256, 0, stream>>>(Xf, qg);

    // K2: Q = Qg @ Wq^T + bq  (exact f32 WMMA)
    proj16_wmma_kernel<<<1, 128, 0, stream>>>(qg, Wq, bq, q);

    // K3: fold Wk + scale into per-head query
    fold_qk_kernel<<<B_ * NH, C_, 0, stream>>>(q, Wk, bk, qk, qb);

    // K4: fused per-window scores/softmax/weighted-sum (403 MB stream, LDS-staged)
    attn_window_kernel<<<dim3(NW, B_), 64, SM_TOT * sizeof(float), stream>>>(Xs, qk, qb, y);

    // K5a: out_heads = Wv_h * y / nW + bv
    vproj_kernel<<<(B_ * C_ + 255) / 256, 256, 0, stream>>>(y, Wv, bv, mid);

    // K5b: fin = mid @ Wo^T + bo  (exact f32 WMMA)
    proj16_wmma_kernel<<<1, 128, 0, stream>>>(mid, Wo, bo, fin);

    // K6: broadcast to (B,C,H,W) (403 MB stream)
    bcast_kernel<<<B_ * C_, 256, 0, stream>>>(fin, out);
}